// bspline_array_64381559767112
// MI455X (gfx1250) — compile-verified
//
#include <hip/hip_runtime.h>

// B-spline (Cox-de Boor, degree 3) batched evaluation for MI455X (gfx1250).
//   x: [B, 64] f32, t: [64, 68] knots, c: [64, 64] coeffs -> out [B, 64]
// Block = 256 threads handles one spline j (blockIdx.y) x 256 batch rows.
// Knot-uniform reciprocal tables live in LDS; per-thread recursion in VGPRs;
// final 64-term dot done with V_WMMA_F32_16X16X4_F32 (f32, no precision loss).

#define DEGREE    3
#define C_DIM     64
#define NSPLINES  64
#define T_DIM     68     // C_DIM + DEGREE + 1
#define NB0       67     // T_DIM - 1 (degree-0 basis count)
#define BLOCK     256

typedef float v2f __attribute__((ext_vector_type(2)));
typedef float v8f __attribute__((ext_vector_type(8)));

#define GAS __attribute__((address_space(1)))
#define LAS __attribute__((address_space(3)))

// ---- CDNA5 async global->LDS staging (ASYNCcnt) --------------------------
__device__ __forceinline__ void async_ld_b32(const float* gsrc, float* ldst) {
#if __has_builtin(__builtin_amdgcn_global_load_async_to_lds_b32)
  // prototype (from diagnostics): (as1 "int"*, as3 "int"*, imm offset, imm cpol)
  __builtin_amdgcn_global_load_async_to_lds_b32(
      (GAS int*)(GAS void*)gsrc, (LAS int*)(LAS void*)ldst, 0, 0);
#else
  unsigned l = (unsigned)(size_t)ldst;            // low 32 bits = LDS address
  unsigned long long g = (unsigned long long)(size_t)gsrc;
  asm volatile("global_load_async_to_lds_b32 %0, %1, off"
               :: "v"(l), "v"(g) : "memory");
#endif
}

__device__ __forceinline__ void wait_async0() {
#if __has_builtin(__builtin_amdgcn_s_wait_asynccnt)
  __builtin_amdgcn_s_wait_asynccnt(0);
#else
  asm volatile("s_wait_asynccnt 0" ::: "memory");
#endif
}

__global__ __launch_bounds__(BLOCK)
void bspline_array_kernel(const float* __restrict__ X,
                          const float* __restrict__ T,
                          const float* __restrict__ Cc,
                          float* __restrict__ O) {
  __shared__ float lt[T_DIM];        // knots of spline j
  __shared__ float lc[C_DIM];        // coefficients of spline j
  __shared__ float inv1[NB0];        // 1/(t[i+1]-t[i]) (0 if den==0)
  __shared__ float inv2[NB0 - 1];    // 1/(t[i+2]-t[i])
  __shared__ float inv3[NB0 - 2];    // 1/(t[i+3]-t[i])

  const int tid = threadIdx.x;
  const int j   = blockIdx.y;

  // ---- stage t-row and c-row into LDS via async tensor path ----
  if (tid < T_DIM) {
    async_ld_b32(&T[j * T_DIM + tid], &lt[tid]);
  } else if (tid < T_DIM + C_DIM) {
    async_ld_b32(&Cc[j * C_DIM + (tid - T_DIM)], &lc[tid - T_DIM]);
  }
  wait_async0();
  __syncthreads();

  // ---- block-uniform safe reciprocals (Cox-de Boor 0/0 := 0) ----
  if (tid < 67) {
    float d = lt[tid + 1] - lt[tid];
    inv1[tid] = (d == 0.0f) ? 0.0f : 1.0f / d;
  } else if (tid < 133) {
    int i = tid - 67;
    float d = lt[i + 2] - lt[i];
    inv2[i] = (d == 0.0f) ? 0.0f : 1.0f / d;
  } else if (tid < 198) {
    int i = tid - 133;
    float d = lt[i + 3] - lt[i];
    inv3[i] = (d == 0.0f) ? 0.0f : 1.0f / d;
  }
  __syncthreads();

  const int   b = blockIdx.x * BLOCK + tid;
  const float x = X[(size_t)b * NSPLINES + j];

  // s[i] = x - t[i]  (left numerator; right numerator = -s[i+k+1])
  float s[T_DIM];
#pragma unroll
  for (int i = 0; i < T_DIM; ++i) s[i] = x - lt[i];

  // degree-0 basis: t[i] <= x < t[i+1]  <=>  s[i] >= 0 && s[i+1] < 0
  float N[NB0];
#pragma unroll
  for (int i = 0; i < NB0; ++i)
    N[i] = (s[i] >= 0.0f && s[i + 1] < 0.0f) ? 1.0f : 0.0f;

  // recursion levels (in-place ascending is safe: uses old N[i], N[i+1])
#pragma unroll
  for (int i = 0; i < 66; ++i)   // k = 1
    N[i] = s[i] * inv1[i] * N[i] - s[i + 2] * inv1[i + 1] * N[i + 1];
#pragma unroll
  for (int i = 0; i < 65; ++i)   // k = 2
    N[i] = s[i] * inv2[i] * N[i] - s[i + 3] * inv2[i + 1] * N[i + 1];
#pragma unroll
  for (int i = 0; i < 64; ++i)   // k = 3
    N[i] = s[i] * inv3[i] * N[i] - s[i + 4] * inv3[i + 1] * N[i + 1];

  // ---- S(x) = sum_i c[i] * N[i]  via V_WMMA_F32_16X16X4_F32 ----
#if __has_builtin(__builtin_amdgcn_wmma_f32_16x16x4_f32)
  // Wave holds 32 batch rows (lane l <-> row b0+l). Two 16-row groups.
  // A (16x4 f32): lanes 0-15 carry K=0,1 in VGPR0/1; lanes 16-31 carry K=2,3.
  // B (4x16): replicate c[k] across all columns => every D column = result.
  const int  lane = tid & 31;
  const bool lo   = lane < 16;
  v8f d0 = {0.f,0.f,0.f,0.f,0.f,0.f,0.f,0.f};
  v8f d1 = d0;
#pragma unroll
  for (int st = 0; st < 16; ++st) {
    const int i0 = 4 * st;
    float n0 = N[i0], n1 = N[i0 + 1], n2 = N[i0 + 2], n3 = N[i0 + 3];
    // half-swap: lane l gets lane (l^16)'s value
    float x0 = __shfl_xor(n0, 16, 32);
    float x1 = __shfl_xor(n1, 16, 32);
    float x2 = __shfl_xor(n2, 16, 32);
    float x3 = __shfl_xor(n3, 16, 32);
    v2f a0;                       // group0: rows b0+0..15
    a0.x = lo ? n0 : x2;          //   lane>=16 supplies row (lane-16) K=2
    a0.y = lo ? n1 : x3;
    v2f a1;                       // group1: rows b0+16..31
    a1.x = lo ? x0 : n2;          //   lane<16 supplies row (lane+16) K=0
    a1.y = lo ? x1 : n3;
    v2f bm;                       // c replicated over the 16 N-columns
    bm.x = lo ? lc[i0]     : lc[i0 + 2];
    bm.y = lo ? lc[i0 + 1] : lc[i0 + 3];
    d0 = __builtin_amdgcn_wmma_f32_16x16x4_f32(false, a0, false, bm,
                                               (short)0, d0, false, false);
    d1 = __builtin_amdgcn_wmma_f32_16x16x4_f32(false, a1, false, bm,
                                               (short)0, d1, false, false);
  }
  // D layout: lane 0 holds rows 0..7 (VGPR r), lane 16 holds rows 8..15.
  const int b0 = blockIdx.x * BLOCK + (tid & ~31);
  if ((lane & 15) == 0) {
    const int rbase = (lane >> 4) * 8;  // 0 for lane 0, 8 for lane 16
#pragma unroll
    for (int r = 0; r < 8; ++r) {
      O[(size_t)(b0 + rbase + r)      * NSPLINES + j] = d0[r];
      O[(size_t)(b0 + 16 + rbase + r) * NSPLINES + j] = d1[r];
    }
  }
#else
  float acc = 0.0f;
#pragma unroll
  for (int i = 0; i < C_DIM; ++i) acc = fmaf(lc[i], N[i], acc);
  O[(size_t)b * NSPLINES + j] = acc;
#endif
}

extern "C" void kernel_launch(void* const* d_in, const int* in_sizes, int n_in,
                              void* d_out, int out_size, void* d_ws, size_t ws_size,
                              hipStream_t stream) {
  (void)n_in; (void)out_size; (void)d_ws; (void)ws_size;
  const float* X  = (const float*)d_in[0];   // [B, 64]
  const float* T  = (const float*)d_in[1];   // [64, 68]
  const float* Cc = (const float*)d_in[2];   // [64, 64]
  float*       O  = (float*)d_out;           // [B, 64]
  const int B = in_sizes[0] / NSPLINES;
  dim3 grid(B / BLOCK, NSPLINES, 1);
  bspline_array_kernel<<<grid, dim3(BLOCK, 1, 1), 0, stream>>>(X, T, Cc, O);
}